// NONA_38019050504216
// MI455X (gfx1250) — compile-verified
//
#include <hip/hip_runtime.h>
#include <hip/hip_bf16.h>
#include <math.h>

#define N_ROWS 4096
#define M_ROWS 8192
#define D_DIM  1024
#define C_DIM  128
#define WAVES  8

typedef float v2f __attribute__((ext_vector_type(2)));
typedef float v8f __attribute__((ext_vector_type(8)));

static __device__ __forceinline__ v8f wmma_f32(v2f a, v2f b, v8f c) {
  // D = A(16x4, f32) * B(4x16, f32) + C(16x16, f32)
  return __builtin_amdgcn_wmma_f32_16x16x4_f32(false, a, false, b, (short)0, c,
                                               false, false);
}

// ---------------------------------------------------------------------------
// Prologue: row squared-norms of x (N rows) and x_n (M rows) into workspace.
// One wave32 per row.
// ---------------------------------------------------------------------------
__global__ __launch_bounds__(256) void nona_norms_kernel(
    const float* __restrict__ x, const float* __restrict__ xn,
    float* __restrict__ xsq, float* __restrict__ xnsq) {
  int gwave = (int)((blockIdx.x * blockDim.x + threadIdx.x) >> 5);
  int lane = (int)(threadIdx.x & 31);
  const float* src;
  float* dst;
  int row;
  if (gwave < N_ROWS) {
    src = x; dst = xsq; row = gwave;
  } else {
    row = gwave - N_ROWS;
    if (row >= M_ROWS) return;
    src = xn; dst = xnsq;
  }
  const float* p = src + (size_t)row * D_DIM;
  float s = 0.f;
  for (int k = lane; k < D_DIM; k += 32) {
    float v = p[k];
    s = fmaf(v, v, s);
  }
#pragma unroll
  for (int off = 16; off > 0; off >>= 1) s += __shfl_xor(s, off, 32);
  if (lane == 0) dst[row] = s;
}

// ---------------------------------------------------------------------------
// Fused kernel: per block = one 16-row tile of x; 8 waves split the M axis and
// run flash-attention-style online softmax; both GEMMs on V_WMMA_F32_16X16X4.
// ---------------------------------------------------------------------------
__global__ __launch_bounds__(256) void nona_fused_kernel(
    const float* __restrict__ x, const float* __restrict__ xn,
    const float* __restrict__ y, const float* __restrict__ xsq,
    const float* __restrict__ xnsq, float* __restrict__ out) {
  extern __shared__ float smem[];
  float* lds_x   = smem;                        // 16*1024 = 16384 floats
  float* lds_xsq = lds_x + 16 * D_DIM;          // 16
  float* lds_p   = lds_xsq + 16;                // 8 waves * 256
  float* lds_m   = lds_p + WAVES * 256;         // 8 * 16
  float* lds_l   = lds_m + WAVES * 16;          // 8 * 16
  float* lds_out = lds_l + WAVES * 16;          // 16 * 128

  const int tid  = (int)threadIdx.x;
  const int lane = tid & 31;
  const int wave = tid >> 5;
  const int half = lane >> 4;   // 0: lanes 0-15, 1: lanes 16-31
  const int l16  = lane & 15;

  const int rowbase = (int)blockIdx.x * 16;

  // Load the 16x1024 x tile into LDS (rows are contiguous in memory).
  {
    const float4* src4 = (const float4*)(x + (size_t)rowbase * D_DIM);
    float4* dst4 = (float4*)lds_x;
    for (int i = tid; i < 16 * D_DIM / 4; i += 256) dst4[i] = src4[i];
  }
  if (tid < 16) lds_xsq[tid] = xsq[rowbase + tid];
  __syncthreads();

  // Per-lane copy of x row norms for this lane's 8 output rows (r + 8*half).
  float rq[8];
#pragma unroll
  for (int r = 0; r < 8; ++r) rq[r] = lds_xsq[8 * half + r];

  // Online-softmax state and output accumulator (16 rows x 128 cols).
  float m_r[8], l_r[8];
#pragma unroll
  for (int r = 0; r < 8; ++r) { m_r[r] = -3.0e38f; l_r[r] = 0.f; }
  v8f oacc[8] = {};

  // A operand base (x tile in LDS): row = l16, k-pair offset 2*half.
  const float* xa = lds_x + l16 * D_DIM + 2 * half;
  float* pl = lds_p + wave * 256;  // 16x16 probability staging, per wave

  // Each wave owns column tiles ct = wave, wave+8, ... (512 tiles / 8 waves).
  for (int ct = wave; ct < M_ROWS / 16; ct += WAVES) {
    const int colbase = ct * 16;

    // ----- GEMM1: S = x_tile . x_n_tile^T over D, fp32 WMMA 16x16x4 -----
    const float* xb = xn + (size_t)(colbase + l16) * D_DIM + 2 * half;
    v8f s = {};
#pragma unroll 4
    for (int k = 0; k < D_DIM; k += 8) {
      v2f a0 = *(const v2f*)(xa + k);
      v2f a1 = *(const v2f*)(xa + k + 4);
      v2f b0 = *(const v2f*)(xb + k);
      v2f b1 = *(const v2f*)(xb + k + 4);
      s = wmma_f32(a0, b0, s);
      s = wmma_f32(a1, b1, s);
    }

    // ----- similarity + online softmax update -----
    const float cq = xnsq[colbase + l16];
    float p[8];
#pragma unroll
    for (int r = 0; r < 8; ++r) {
      float d2 = fmaxf(rq[r] + cq - 2.0f * s[r], 0.0f);
      p[r] = -sqrtf(d2);  // sim value
    }
#pragma unroll
    for (int r = 0; r < 8; ++r) {
      // row max over the 16 lanes of this half-group
      float mx = p[r];
#pragma unroll
      for (int off = 1; off < 16; off <<= 1) mx = fmaxf(mx, __shfl_xor(mx, off, 32));
      float mnew = fmaxf(m_r[r], mx);
      float scale = __expf(m_r[r] - mnew);
      float pv = __expf(p[r] - mnew);
      float rs = pv;
#pragma unroll
      for (int off = 1; off < 16; off <<= 1) rs += __shfl_xor(rs, off, 32);
      l_r[r] = l_r[r] * scale + rs;
      m_r[r] = mnew;
#pragma unroll
      for (int t = 0; t < 8; ++t) oacc[t][r] *= scale;
      p[r] = pv;
    }

    // Stage probabilities in LDS to transpose C-layout -> A-layout.
#pragma unroll
    for (int r = 0; r < 8; ++r) pl[(r + 8 * half) * 16 + l16] = p[r];

    // ----- GEMM2: oacc += P(16x16) . y_tile(16x128), fp32 WMMA 16x16x4 -----
    const float* yb = y + (size_t)(colbase + 2 * half) * C_DIM + l16;
#pragma unroll
    for (int kk = 0; kk < 16; kk += 4) {
      v2f a = *(const v2f*)(pl + l16 * 16 + kk + 2 * half);
#pragma unroll
      for (int t = 0; t < 8; ++t) {
        v2f b;
        b.x = yb[(size_t)kk * C_DIM + 16 * t];
        b.y = yb[(size_t)(kk + 1) * C_DIM + 16 * t];
        oacc[t] = wmma_f32(a, b, oacc[t]);
      }
    }
  }

  // ----- combine the 8 per-wave partial softmaxes (flash-attention merge) ---
  if (l16 == 0) {
#pragma unroll
    for (int r = 0; r < 8; ++r) {
      lds_m[wave * 16 + r + 8 * half] = m_r[r];
      lds_l[wave * 16 + r + 8 * half] = l_r[r];
    }
  }
  for (int i = tid; i < 16 * C_DIM; i += 256) lds_out[i] = 0.f;
  __syncthreads();

  float g[8];
#pragma unroll
  for (int r = 0; r < 8; ++r) {
    const int row = r + 8 * half;
    float mx = lds_m[row];
#pragma unroll
    for (int w = 1; w < WAVES; ++w) mx = fmaxf(mx, lds_m[w * 16 + row]);
    float L = 0.f;
#pragma unroll
    for (int w = 0; w < WAVES; ++w)
      L += lds_l[w * 16 + row] * __expf(lds_m[w * 16 + row] - mx);
    g[r] = __expf(m_r[r] - mx) / L;  // own wave's rescale, division folded in
  }

#pragma unroll
  for (int t = 0; t < 8; ++t) {
#pragma unroll
    for (int r = 0; r < 8; ++r) {
      atomicAdd(&lds_out[(r + 8 * half) * C_DIM + 16 * t + l16], oacc[t][r] * g[r]);
    }
  }
  __syncthreads();

  float* op = out + (size_t)rowbase * C_DIM;
  for (int i = tid; i < 16 * C_DIM; i += 256) op[i] = lds_out[i];
}

// ---------------------------------------------------------------------------
extern "C" void kernel_launch(void* const* d_in, const int* in_sizes, int n_in,
                              void* d_out, int out_size, void* d_ws, size_t ws_size,
                              hipStream_t stream) {
  (void)in_sizes; (void)n_in; (void)out_size; (void)ws_size;
  const float* x  = (const float*)d_in[0];
  const float* xn = (const float*)d_in[1];
  const float* y  = (const float*)d_in[2];
  // d_in[3] = log_T, unused by the reference math (T computed then discarded)
  float* out = (float*)d_out;

  float* xsq  = (float*)d_ws;       // N_ROWS floats
  float* xnsq = xsq + N_ROWS;       // M_ROWS floats

  // Prologue: one wave per row; (N+M) waves / 8 waves-per-block.
  nona_norms_kernel<<<(N_ROWS + M_ROWS) / 8, 256, 0, stream>>>(x, xn, xsq, xnsq);

  // Main fused kernel: one block per 16-row tile of x.
  const size_t lds_bytes =
      (16 * D_DIM + 16 + WAVES * 256 + WAVES * 16 + WAVES * 16 + 16 * C_DIM) *
      sizeof(float);  // ~83 KB of the 320 KB WGP LDS
  nona_fused_kernel<<<N_ROWS / 16, 256, lds_bytes, stream>>>(x, xn, y, xsq, xnsq, out);
}